// HeteroGraphSAGEEncoder_33681133535938
// MI455X (gfx1250) — compile-verified
//
#include <hip/hip_runtime.h>
#include <hip/hip_bf16.h>

#define ND     40000
#define NPROT  15000
#define E_DDI  600000
#define E_TGT  250000

#define KC      32          // K-chunk staged in LDS (dwords)
#define WSTRIDE 36          // LDS row stride (words) = KC + 4 pad -> conflict-free b64 reads

typedef __attribute__((ext_vector_type(2))) float v2f;
typedef __attribute__((ext_vector_type(8))) float v8f;
typedef __attribute__((ext_vector_type(4))) unsigned int v4u;
typedef __attribute__((ext_vector_type(4))) int v4i;
typedef __attribute__((ext_vector_type(8))) int v8i;

static __device__ __forceinline__ v8f wmma4(v2f a, v2f b, v8f c) {
    return __builtin_amdgcn_wmma_f32_16x16x4_f32(false, a, false, b, (short)0, c, false, false);
}

// ---------------------------------------------------------------------------
// TDM: DMA one W tile (64 rows x KC dwords, row stride K dwords) -> LDS.
// LDS padding: after every 32 dwords insert 4 dwords  => row stride 36 words.
// OOB columns (x >= remK) read as zero => K-tail / K=5 pad handled in HW.
// D# layout per CDNA5 ISA 8.3/8.4 (group0: count/lds/global/type,
// group1: data_size, pad, tensor dims, tile dims, dim0 stride).
// ---------------------------------------------------------------------------
static __device__ __forceinline__ void tdm_load_tile(unsigned lds_off, const float* gptr,
                                                     int remK, int remN, int strideK)
{
    unsigned long long ga = (unsigned long long)(size_t)gptr;
    v4u g0;
    g0[0] = 1u;                                            // count=1 (valid user D#)
    g0[1] = lds_off;                                       // lds_addr (bytes)
    g0[2] = (unsigned)(ga & 0xffffffffu);                  // global_addr[31:0] (tile start)
    g0[3] = (unsigned)((ga >> 32) & 0x01ffffffu)           // global_addr[56:32]
          | (2u << 30);                                    // type = 2 ("image")
    v8i g1;
    g1[0] = (2 << 16)                                      // data_size = 4 bytes
          | (1 << 20)                                      // pad_enable
          | (4 << 22)                                      // pad_interval: 32 dwords
          | (3 << 25);                                     // pad_amount: 4 dwords
    g1[1] = (remK & 0xffff) << 16;                         // tensor_dim0[15:0]
    g1[2] = ((remK >> 16) & 0xffff)                        // tensor_dim0[31:16]
          | ((remN & 0xffff) << 16);                       // tensor_dim1[15:0]
    g1[3] = ((remN >> 16) & 0xffff)                        // tensor_dim1[31:16]
          | (KC << 16);                                    // tile_dim0 = 32 dwords
    g1[4] = 64;                                            // tile_dim1 = 64 rows, tile_dim2 = 0
    g1[5] = strideK;                                       // tensor_dim0_stride[31:0]
    g1[6] = 0;                                             // stride0[47:32], stride1[15:0]
    g1[7] = 0;                                             // stride1[47:16] (unused, 2D)
    v4i z4 = (v4i)0;
#if __clang_major__ >= 23
    v8i z8 = (v8i)0;
    __builtin_amdgcn_tensor_load_to_lds(g0, g1, z4, z4, z8, 0);
#else
    __builtin_amdgcn_tensor_load_to_lds(g0, g1, z4, z4, 0);
#endif
}

// ---------------------------------------------------------------------------
// GEMM: C[M,N] = X[M,K] @ W[N,K]^T   (fp32, V_WMMA_F32_16X16X4_F32)
// Block: 256 threads = 8 waves; wave tile = 32(M) x 64(N) -> 8 accumulators.
// W tile double-buffered in LDS via Tensor Data Mover (wave 0 issues DMA,
// TENSORcnt in-order completion => wait(1) overlaps next DMA with compute).
// Per k-step/wave: 2 global_load_b64 (A) + 4 ds_load_b64 (B) + 8 WMMA.
// K multiple of 4 (980 / 256 / 8-padded); N multiple of 64 (256 / 64).
// ---------------------------------------------------------------------------
__global__ __launch_bounds__(256)
void gemm_wmma_f32(const float* __restrict__ X, const float* __restrict__ Wt,
                   float* __restrict__ C, int M, int K, int N)
{
    __shared__ float wtile[2][64 * WSTRIDE];

    const int tid   = threadIdx.x;
    const int lane  = tid & 31;
    const int wv    = tid >> 5;
    const int l15   = lane & 15;
    const int hi    = lane >> 4;        // 0: lanes 0-15, 1: lanes 16-31
    const int khalf = hi * 2;           // k-pair offset within WMMA fragment
    const int m0    = blockIdx.x * 256 + wv * 32;
    const int n0    = blockIdx.y * 64;

    int r0 = m0 + l15;       if (r0 >= M) r0 = M - 1;
    int r1 = m0 + 16 + l15;  if (r1 >= M) r1 = M - 1;
    const float* pa0 = X + (size_t)r0 * K + khalf;
    const float* pa1 = X + (size_t)r1 * K + khalf;

    // low 32 bits of a generic LDS pointer = LDS byte offset
    const unsigned ldsbase = (unsigned)(size_t)(&wtile[0][0]);
    const float* wrow = Wt + (size_t)n0 * K;   // tile row origin (64 rows)
    const int remN = N - n0;

    const int nch = (K + KC - 1) / KC;

    v8f accA[4] = {{}, {}, {}, {}};
    v8f accB[4] = {{}, {}, {}, {}};

    if (wv == 0)
        tdm_load_tile(ldsbase, wrow, K, remN, K);          // chunk 0 -> buf 0

    int p = 0;
    for (int c = 0; c < nch; ++c) {
        const int kc = c * KC;

        if (wv == 0) {
            if (c + 1 < nch) {                              // prefetch chunk c+1 -> buf p^1
                tdm_load_tile(ldsbase + (unsigned)(p ^ 1) * (64 * WSTRIDE * 4),
                              wrow + kc + KC, K - (kc + KC), remN, K);
                __builtin_amdgcn_s_wait_tensorcnt(1);       // buf p complete (in-order)
            } else {
                __builtin_amdgcn_s_wait_tensorcnt(0);
            }
        }
        if (c + 1 < nch) {                                  // next A chunk -> global_prefetch_b8
            __builtin_prefetch(pa0 + kc + KC, 0, 3);
            __builtin_prefetch(pa1 + kc + KC, 0, 3);
        }
        __syncthreads();

        const float* bb = &wtile[p][(size_t)l15 * WSTRIDE + khalf];
        const int kend = (K - kc < KC) ? (K - kc) : KC;
        if (kend == KC) {
            #pragma unroll
            for (int kk = 0; kk < KC; kk += 4) {
                v2f a0 = *(const v2f*)(pa0 + kc + kk);
                v2f a1 = *(const v2f*)(pa1 + kc + kk);
                #pragma unroll
                for (int t = 0; t < 4; ++t) {
                    v2f b = *(const v2f*)(bb + t * (16 * WSTRIDE) + kk);
                    accA[t] = wmma4(a0, b, accA[t]);
                    accB[t] = wmma4(a1, b, accB[t]);
                }
            }
        } else {
            for (int kk = 0; kk < kend; kk += 4) {          // K-tail: A stays in-bounds
                v2f a0 = *(const v2f*)(pa0 + kc + kk);
                v2f a1 = *(const v2f*)(pa1 + kc + kk);
                #pragma unroll
                for (int t = 0; t < 4; ++t) {
                    v2f b = *(const v2f*)(bb + t * (16 * WSTRIDE) + kk);
                    accA[t] = wmma4(a0, b, accA[t]);
                    accB[t] = wmma4(a1, b, accB[t]);
                }
            }
        }

        __syncthreads();
        p ^= 1;
    }

    // store: D layout -> vgpr v: lanes<16 M=v, lanes>=16 M=v+8; N = n0+16t+l15
    if (m0 < M) {
        const int mb = m0 + hi * 8;
        #pragma unroll
        for (int v = 0; v < 8; ++v) {
            const int mrA = mb + v;
            if (mrA < M) {
                size_t o = (size_t)mrA * N + n0 + l15;
                #pragma unroll
                for (int t = 0; t < 4; ++t) C[o + 16 * t] = accA[t][v];
            }
            const int mrB = mb + 16 + v;
            if (mrB < M) {
                size_t o = (size_t)mrB * N + n0 + l15;
                #pragma unroll
                for (int t = 0; t < 4; ++t) C[o + 16 * t] = accB[t][v];
            }
        }
    }
}

// ---------------------------------------------------------------------------
// Edge scatter: agg[dst[e]] += msg[src[e]] * inv[dst[e]]   (4 floats / thread)
// ---------------------------------------------------------------------------
__global__ __launch_bounds__(256)
void scatter_scaled(const float* __restrict__ msg, const int* __restrict__ src,
                    const int* __restrict__ dst, const float* __restrict__ inv,
                    float* __restrict__ agg, long long nwork, int cshift, int Ho)
{
    long long i = (long long)blockIdx.x * blockDim.x + threadIdx.x;
    if (i >= nwork) return;
    int e = (int)(i >> cshift);
    int c = ((int)i & ((1 << cshift) - 1)) << 2;
    int s = src[e], d = dst[e];
    float sc = inv[d];
    const float4 v = *(const float4*)(msg + (size_t)s * Ho + c);
    float* p = agg + (size_t)d * Ho + c;
    unsafeAtomicAdd(p + 0, v.x * sc);
    unsafeAtomicAdd(p + 1, v.y * sc);
    unsafeAtomicAdd(p + 2, v.z * sc);
    unsafeAtomicAdd(p + 3, v.w * sc);
}

__global__ void zero_f32(float* __restrict__ p, long long n)
{
    long long i = (long long)blockIdx.x * blockDim.x + threadIdx.x;
    if (i < n) p[i] = 0.f;
}

__global__ void count_edges(const int* __restrict__ idx, float* __restrict__ cnt, int n)
{
    int i = blockIdx.x * blockDim.x + threadIdx.x;
    if (i < n) unsafeAtomicAdd(&cnt[idx[i]], 1.0f);
}

__global__ void invert_counts(float* __restrict__ c, int n)
{
    int i = blockIdx.x * blockDim.x + threadIdx.x;
    if (i < n) c[i] = 1.0f / fmaxf(c[i], 1.0f);
}

__global__ void pad_rows_kernel(const float* __restrict__ in, float* __restrict__ out,
                                int rows, int K, int Kp)
{
    int i = blockIdx.x * blockDim.x + threadIdx.x;
    if (i >= rows * Kp) return;
    int r = i / Kp, k = i - r * Kp;
    out[i] = (k < K) ? in[r * K + k] : 0.f;
}

__global__ void add_mat(const float* __restrict__ a, const float* __restrict__ b,
                        float* __restrict__ o, long long n)
{
    long long i = (long long)blockIdx.x * blockDim.x + threadIdx.x;
    if (i < n) o[i] = a[i] + b[i];
}

__global__ void finalize_kernel(const float* __restrict__ acc, const float* __restrict__ b1,
                                const float* __restrict__ b2, float* __restrict__ out,
                                long long n, int hmask, int relu)
{
    long long i = (long long)blockIdx.x * blockDim.x + threadIdx.x;
    if (i >= n) return;
    int col = (int)i & hmask;
    float v = acc[i] + b1[col] + (b2 ? b2[col] : 0.f);
    if (relu) v = fmaxf(v, 0.f);
    out[i] = v;
}

extern "C" void kernel_launch(void* const* d_in, const int* in_sizes, int n_in,
                              void* d_out, int out_size, void* d_ws, size_t ws_size,
                              hipStream_t stream)
{
    (void)in_sizes; (void)n_in; (void)out_size; (void)ws_size;

    const float* x_drug  = (const float*)d_in[0];
    const float* x_prot  = (const float*)d_in[1];
    const int*   ddi_s   = (const int*)d_in[2];
    const int*   ddi_d   = ddi_s + E_DDI;
    const int*   tgt_src = (const int*)d_in[3];
    const int*   tgt_dst = (const int*)d_in[4];
    const float* P[27];
    for (int i = 0; i < 27; ++i) P[i] = (const float*)d_in[5 + i];

    // ---- workspace layout (all fp32, 16B aligned) ----
    float* wsf = (float*)d_ws;
    size_t off = 0;
    auto take = [&](size_t n) { float* p = wsf + off; off += (n + 3) & ~(size_t)3; return p; };
    float* bufD0 = take((size_t)ND * 256);
    float* bufD1 = take((size_t)ND * 256);
    float* bufP0 = take((size_t)NPROT * 256);
    float* bufP1 = take((size_t)NPROT * 256);
    float* msgD  = take((size_t)ND * 256);      // reused for ddi and tgt messages
    float* msgP  = take((size_t)NPROT * 256);
    float* aggD  = take((size_t)ND * 256);      // self-GEMM writes here, scatters add
    float* aggP  = take((size_t)NPROT * 256);
    float* invD  = take(ND);
    float* invR  = take(ND);
    float* invP  = take(NPROT);
    float* xpPad = take((size_t)NPROT * 8);     // x_prot padded K: 5 -> 8
    float* wPadT = take(256 * 8);               // l1_tgt_Wr (256x5 -> 256x8)
    float* wPadR = take(256 * 8);               // l1_rev_Wl (256x5 -> 256x8)
    float* wSum  = take(256 * 980);             // dWr + rWr

    auto blk = [](long long n) { return (unsigned)((n + 255) / 256); };
    auto gemm = [&](const float* X, const float* Wt, float* C, int M, int K, int N) {
        dim3 g((unsigned)((M + 255) / 256), (unsigned)(N / 64));
        gemm_wmma_f32<<<g, 256, 0, stream>>>(X, Wt, C, M, K, N);
    };

    // ---- inverse mean-counts (edge lists are layer-invariant) ----
    zero_f32<<<blk(ND), 256, 0, stream>>>(invD, ND);
    zero_f32<<<blk(ND), 256, 0, stream>>>(invR, ND);
    zero_f32<<<blk(NPROT), 256, 0, stream>>>(invP, NPROT);
    count_edges<<<blk(E_DDI), 256, 0, stream>>>(ddi_d, invD, E_DDI);
    count_edges<<<blk(E_TGT), 256, 0, stream>>>(tgt_src, invR, E_TGT);
    count_edges<<<blk(E_TGT), 256, 0, stream>>>(tgt_dst, invP, E_TGT);
    invert_counts<<<blk(ND), 256, 0, stream>>>(invD, ND);
    invert_counts<<<blk(ND), 256, 0, stream>>>(invR, ND);
    invert_counts<<<blk(NPROT), 256, 0, stream>>>(invP, NPROT);

    // ---- pad K=5 operands to K=8 (layer 1 only) ----
    pad_rows_kernel<<<blk((long long)NPROT * 8), 256, 0, stream>>>(x_prot, xpPad, NPROT, 5, 8);
    pad_rows_kernel<<<blk(256 * 8), 256, 0, stream>>>(P[5], wPadT, 256, 5, 8);  // l1_tgt_Wr
    pad_rows_kernel<<<blk(256 * 8), 256, 0, stream>>>(P[6], wPadR, 256, 5, 8);  // l1_rev_Wl

    const float* xd_in = x_drug;
    const float* xp_in = xpPad;
    float* out = (float*)d_out;

    for (int l = 0; l < 3; ++l) {
        const float* dWl = P[9 * l + 0]; const float* db = P[9 * l + 1]; const float* dWr = P[9 * l + 2];
        const float* tWl = P[9 * l + 3]; const float* tb = P[9 * l + 4]; const float* tWr = P[9 * l + 5];
        const float* rWl = P[9 * l + 6]; const float* rb = P[9 * l + 7]; const float* rWr = P[9 * l + 8];
        const int Kd = (l == 0) ? 980 : 256;
        const int Kp = (l == 0) ? 8   : 256;
        const int Ho = (l == 2) ? 64  : 256;
        if (l == 0) { tWr = wPadT; rWl = wPadR; }
        float* xd_out = (l == 0) ? bufD0 : (l == 1) ? bufD1 : out;
        float* xp_out = (l == 0) ? bufP0 : (l == 1) ? bufP1 : out + (size_t)ND * 64;
        const int relu   = (l < 2);
        const int cshift = (Ho == 256) ? 6 : 4;    // Ho/4 threads per edge, 4 columns each
        const long long nwD = (long long)E_DDI << cshift;
        const long long nwT = (long long)E_TGT << cshift;

        // drug output: self term xd@(dWr+rWr)^T written straight into aggD,
        // then both mean-aggregations atomically added (pre-scaled by 1/cnt).
        add_mat<<<blk((long long)Ho * Kd), 256, 0, stream>>>(dWr, rWr, wSum, (long long)Ho * Kd);
        gemm(xd_in, wSum, aggD, ND, Kd, Ho);
        gemm(xd_in, dWl, msgD, ND, Kd, Ho);
        scatter_scaled<<<blk(nwD), 256, 0, stream>>>(msgD, ddi_s, ddi_d, invD, aggD, nwD, cshift, Ho);
        gemm(xp_in, rWl, msgP, NPROT, Kp, Ho);
        scatter_scaled<<<blk(nwT), 256, 0, stream>>>(msgP, tgt_dst, tgt_src, invR, aggD, nwT, cshift, Ho);
        finalize_kernel<<<blk((long long)ND * Ho), 256, 0, stream>>>(
            aggD, db, rb, xd_out, (long long)ND * Ho, Ho - 1, relu);

        // protein output: self term xp@tWr^T into aggP, then tgt aggregation.
        gemm(xp_in, tWr, aggP, NPROT, Kp, Ho);
        gemm(xd_in, tWl, msgD, ND, Kd, Ho);
        scatter_scaled<<<blk(nwT), 256, 0, stream>>>(msgD, tgt_src, tgt_dst, invP, aggP, nwT, cshift, Ho);
        finalize_kernel<<<blk((long long)NPROT * Ho), 256, 0, stream>>>(
            aggP, tb, nullptr, xp_out, (long long)NPROT * Ho, Ho - 1, relu);

        xd_in = xd_out;
        xp_in = xp_out;
    }
}